// DivergenceFreeMLP_24343874633972
// MI455X (gfx1250) — compile-verified
//
#include <hip/hip_runtime.h>
#include <hip/hip_bf16.h>

// ---------------------------------------------------------------------------
// DivergenceFreeMLP: out[b] = (J z - J^T z) . Wp + bp, reduced algebraically:
//   forward piece : ta2 . wv   with wv = Wo @ Wp   (128)
//   backward piece: gh1 . u    with u  = W1^T @ Wp (256)
// Heavy GEMMs run on V_WMMA_F32_16X16X4_F32. All B operands are pre-packed
// into pair-interleaved float2 layout so each B fragment is ONE b64 load,
// and all fragment streams walk pointers with constant strides.
// ---------------------------------------------------------------------------

typedef float v2f __attribute__((ext_vector_type(2)));
typedef float v8f __attribute__((ext_vector_type(8)));

static __device__ __forceinline__ v8f wmma_k4(v2f a, v2f b, v8f c) {
  // (neg_a, A, neg_b, B, c_mod, C, reuse_a, reuse_b)
  return __builtin_amdgcn_wmma_f32_16x16x4_f32(false, a, false, b, (short)0, c,
                                               false, false);
}

static __device__ __forceinline__ v2f ldp(const float2* __restrict__ p) {
  const float2 v = *p;
  v2f r;
  r.x = v.x;
  r.y = v.y;
  return r;
}

static __device__ __forceinline__ v2f lda(const float* __restrict__ p) {
  v2f r;
  r.x = p[0];
  r.y = p[1];
  return r;
}

// ---------------------------------------------------------------------------
// Phase 0a: wv[128] = Wo @ Wp ; u[256] = W1^T @ Wp
// ---------------------------------------------------------------------------
__global__ void dfmlp_vec_kernel(const float* __restrict__ W1,
                                 const float* __restrict__ Wo,
                                 const float* __restrict__ Wp,
                                 float* __restrict__ wv,
                                 float* __restrict__ u) {
  const int t = blockIdx.x * blockDim.x + threadIdx.x;
  if (t < 128) {
    const float* row = Wo + (size_t)t * 2048;
    float s = 0.f;
    for (int j = 0; j < 2048; ++j) s = fmaf(row[j], Wp[j], s);
    wv[t] = s;
  } else if (t < 384) {
    const int l = t - 128;
    float s = 0.f;
    for (int i = 0; i < 2048; ++i) s = fmaf(W1[(size_t)i * 256 + l], Wp[i], s);
    u[l] = s;
  }
}

// ---------------------------------------------------------------------------
// Phase 0b: pair-pack all B operands (transposes folded in):
//   W1P [1024][256] <- W1 rows {2p,2p+1}
//   WoTP[1024][128] <- Wo^T rows {2p,2p+1}  (= Wo cols)
//   W2P [ 128][128] <- W2 rows {2p,2p+1}
//   W2TP[  64][256] <- W2^T rows {2p,2p+1}  (= W2 cols)
// ---------------------------------------------------------------------------
__global__ void dfmlp_pack_kernel(const float* __restrict__ W1,
                                  const float* __restrict__ W2,
                                  const float* __restrict__ Wo,
                                  float2* __restrict__ W1P,
                                  float2* __restrict__ WoTP,
                                  float2* __restrict__ W2P,
                                  float2* __restrict__ W2TP) {
  const int t = blockIdx.x * blockDim.x + threadIdx.x;
  if (t < 262144) {  // 1024 x 256
    const int p = t >> 8, n = t & 255;
    W1P[t] = make_float2(W1[(size_t)(2 * p) * 256 + n],
                         W1[(size_t)(2 * p + 1) * 256 + n]);
  } else if (t < 262144 + 131072) {  // 1024 x 128
    const int i = t - 262144;
    const int p = i >> 7, n = i & 127;
    const float* c = Wo + (size_t)n * 2048 + 2 * p;
    WoTP[i] = make_float2(c[0], c[1]);
  } else if (t < 262144 + 131072 + 16384) {  // 128 x 128
    const int i = t - (262144 + 131072);
    const int p = i >> 7, n = i & 127;
    W2P[i] = make_float2(W2[(size_t)(2 * p) * 128 + n],
                         W2[(size_t)(2 * p + 1) * 128 + n]);
  } else if (t < 262144 + 131072 + 16384 + 16384) {  // 64 x 256
    const int i = t - (262144 + 131072 + 16384);
    const int p = i >> 8, n = i & 255;
    const float* c = W2 + (size_t)n * 128 + 2 * p;
    W2TP[i] = make_float2(c[0], c[1]);
  }
}

// ---------------------------------------------------------------------------
// Phase 1: main fused kernel. 512 blocks x 16 rows; 256 threads = 8 waves.
// ---------------------------------------------------------------------------
#define LDZ 516   // 512 + 4 pad  (bank-conflict-free A-frag reads)
#define LD1 260   // 256 + 4 pad
#define LD2 132   // 128 + 4 pad

__global__ __launch_bounds__(256) void dfmlp_main_kernel(
    const float* __restrict__ x, const float2* __restrict__ W1P,
    const float* __restrict__ b1, const float2* __restrict__ W2P,
    const float* __restrict__ b2, const float* __restrict__ bp,
    const float2* __restrict__ WoTP, const float2* __restrict__ W2TP,
    const float* __restrict__ wv, const float* __restrict__ u,
    float* __restrict__ out) {
  // Overlaid LDS: Zs (stage A, 16*516=8256 f) aliases the stage-B+ arrays,
  // which are barrier-separated from the last Zs read.
  __shared__ float smem[12576];  // 50,304 bytes
  float* Zs = smem;              // [16][LDZ]   stage A only
  float* A1s = smem;             // [16][LD1]   0     .. 4159
  float* TA1s = smem + 4160;     // [16][LD1]   4160  .. 8319
  float* G2s = smem + 8320;      // [16][LD2]   8320  .. 10431
  float* GH2s = smem + 10432;    // [16][LD2]   10432 .. 12543
  float* sfRow = smem + 12544;   // [16]
  float* sbRow = smem + 12560;   // [16]

  const int tid = threadIdx.x;
  const int lane = tid & 31;
  const int wave = tid >> 5;  // 0..7
  const int m0 = blockIdx.x << 4;

  // Row accumulators live past the Zs region -> safe to init up-front.
  if (tid < 16) {
    sfRow[tid] = 0.f;
    sbRow[tid] = 0.f;
  }

  const int n0a = wave << 5;      // W1 / GA1 tile 0 (stride-32 pair)
  const int n0b = n0a + 16;       // W1 / GA1 tile 1
  const int n0g = wave << 4;      // Wo^T / W2 tile
  const int n = lane & 15;
  const int hi = lane >> 4;       // lane group: K pair select
  const int mrow = lane & 15;     // A-frag row
  const int koff = hi << 1;       // A-frag K offset {0,2}

  // ---- Stage A: H1 = z@W1 (two N-tiles) and G2 = z@Wo^T, K=2048 in 4 chunks.
  v8f accA = {0.f, 0.f, 0.f, 0.f, 0.f, 0.f, 0.f, 0.f};
  v8f accB = accA, accG = accA;
  const float* pZbase = Zs + mrow * LDZ + koff;
  for (int c = 0; c < 4; ++c) {
    __syncthreads();  // previous chunk fully consumed
    // cooperative load: 16 rows x 512 cols of this K-chunk into LDS
    for (int i = 0; i < 8; ++i) {
      const int idx = tid + (i << 8);  // 0..2047 float4 slots
      const int row = idx >> 7;        // 128 float4 per row
      const int c4 = idx & 127;
      float4 v = *(const float4*)(x + (size_t)(m0 + row) * 2048 + (c << 9) +
                                  (c4 << 2));
      *(float4*)&Zs[row * LDZ + (c4 << 2)] = v;
    }
    __syncthreads();
    // per-lane fragment stream pointers for this K-chunk
    const float2* pBa = W1P + (size_t)((c << 8) + hi) * 256 + n0a + n;
    const float2* pBb = pBa + 16;
    const float2* pBg = WoTP + (size_t)((c << 8) + hi) * 128 + n0g + n;
#pragma unroll 4
    for (int k0 = 0; k0 < 512; k0 += 4) {
      v2f a = lda(pZbase + k0);        // ds offsets fold into immediates
      v2f bA = ldp(pBa);
      v2f bB = ldp(pBb);
      v2f bG = ldp(pBg);
      pBa += 512;                      // 2 packed rows of 256
      pBb += 512;
      pBg += 256;                      // 2 packed rows of 128
      accA = wmma_k4(a, bA, accA);
      accB = wmma_k4(a, bB, accB);
      accG = wmma_k4(a, bG, accG);
    }
  }
  __syncthreads();  // all Zs reads done before overlay is written

  // ---- Stage B: ReLU / tangent mask; spill a1, ta1, raw ga2 to LDS.
  const float b1a = b1[n0a + n];
  const float b1b = b1[n0b + n];
#pragma unroll
  for (int r = 0; r < 8; ++r) {
    const int m = r + (hi << 3);  // D-layout: lanes16-31 hold rows 8..15
    const float ha = accA[r] + b1a;
    A1s[m * LD1 + n0a + n] = fmaxf(ha, 0.f);
    TA1s[m * LD1 + n0a + n] = (ha > 0.f) ? accA[r] : 0.f;
    const float hb = accB[r] + b1b;
    A1s[m * LD1 + n0b + n] = fmaxf(hb, 0.f);
    TA1s[m * LD1 + n0b + n] = (hb > 0.f) ? accB[r] : 0.f;
    G2s[m * LD2 + n0g + n] = accG[r];
  }
  __syncthreads();

  // ---- Stage C: H2 = a1@W2, T2 = ta1@W2 (K=256); mask; sf partial.
  v8f accH2 = {0.f, 0.f, 0.f, 0.f, 0.f, 0.f, 0.f, 0.f};
  v8f accT2 = accH2;
  {
    const float* pAH = A1s + mrow * LD1 + koff;
    const float* pAT = TA1s + mrow * LD1 + koff;
    const float2* pBW = W2P + (size_t)hi * 128 + n0g + n;
#pragma unroll 4
    for (int k0 = 0; k0 < 256; k0 += 4) {
      v2f aH = lda(pAH + k0);
      v2f aT = lda(pAT + k0);
      v2f bW = ldp(pBW);  // B fragment shared by both WMMAs
      pBW += 256;         // 2 packed rows of 128
      accH2 = wmma_k4(aH, bW, accH2);
      accT2 = wmma_k4(aT, bW, accT2);
    }
  }
  {
    const float b2c = b2[n0g + n];
    const float wvc = wv[n0g + n];
#pragma unroll
    for (int r = 0; r < 8; ++r) {
      const int m = r + (hi << 3);
      const bool m2 = (accH2[r] + b2c) > 0.f;
      const float ta2 = m2 ? accT2[r] : 0.f;
      GH2s[m * LD2 + n0g + n] = m2 ? G2s[m * LD2 + n0g + n] : 0.f;
      atomicAdd(&sfRow[m], ta2 * wvc);  // ds_add_f32
    }
  }
  __syncthreads();

  // ---- Stage D: GA1 = gh2 @ W2^T (K=128, two N-tiles matching stage A's),
  // re-mask with m1 from the retained H1 accumulators, sb partial.
  v8f accGA = {0.f, 0.f, 0.f, 0.f, 0.f, 0.f, 0.f, 0.f};
  v8f accGB = accGA;
  {
    const float* pAG = GH2s + mrow * LD2 + koff;
    const float2* pBa = W2TP + (size_t)hi * 256 + n0a + n;
    const float2* pBb = pBa + 16;
#pragma unroll 4
    for (int k0 = 0; k0 < 128; k0 += 4) {
      v2f aG = lda(pAG + k0);
      v2f bA = ldp(pBa);
      v2f bB = ldp(pBb);
      pBa += 512;  // 2 packed rows of 256
      pBb += 512;
      accGA = wmma_k4(aG, bA, accGA);
      accGB = wmma_k4(aG, bB, accGB);
    }
  }
  {
    const float ua = u[n0a + n];
    const float ub = u[n0b + n];
#pragma unroll
    for (int r = 0; r < 8; ++r) {
      const int m = r + (hi << 3);
      const float g1 = ((accA[r] + b1a) > 0.f) ? accGA[r] : 0.f;
      const float g2 = ((accB[r] + b1b) > 0.f) ? accGB[r] : 0.f;
      atomicAdd(&sbRow[m], fmaf(g1, ua, g2 * ub));
    }
  }
  __syncthreads();

  // ---- Output: v.Wp + bp = sf - sb + bp
  if (tid < 16) out[m0 + tid] = sfRow[tid] - sbRow[tid] + bp[0];
}

// ---------------------------------------------------------------------------
extern "C" void kernel_launch(void* const* d_in, const int* in_sizes, int n_in,
                              void* d_out, int out_size, void* d_ws,
                              size_t ws_size, hipStream_t stream) {
  const float* x = (const float*)d_in[0];   // [8192, 64, 32]
  const float* W1 = (const float*)d_in[1];  // [2048, 256]
  const float* b1 = (const float*)d_in[2];  // [256]
  const float* W2 = (const float*)d_in[3];  // [256, 128]
  const float* b2 = (const float*)d_in[4];  // [128]
  const float* Wo = (const float*)d_in[5];  // [128, 2048]
  // d_in[6] = bo, unused: biases have zero Jacobian contribution.
  const float* Wp = (const float*)d_in[7];  // [2048]
  const float* bp = (const float*)d_in[8];  // [1]
  float* out = (float*)d_out;               // [8192]

  float* ws = (float*)d_ws;
  float* wv = ws;             // 128 floats
  float* u = ws + 128;        // 256 floats
  float2* W1P = (float2*)(ws + 384);   // 262144 pairs (2 MB)
  float2* WoTP = W1P + 262144;         // 131072 pairs (1 MB)
  float2* W2P = WoTP + 131072;         // 16384 pairs
  float2* W2TP = W2P + 16384;          // 16384 pairs
  // total ws: ~3.41 MB

  dfmlp_vec_kernel<<<2, 256, 0, stream>>>(W1, Wo, Wp, wv, u);
  dfmlp_pack_kernel<<<1664, 256, 0, stream>>>(W1, W2, Wo, W1P, WoTP, W2P,
                                              W2TP);
  dfmlp_main_kernel<<<512, 256, 0, stream>>>(x, W1P, b1, W2P, b2, bp, WoTP,
                                             W2TP, wv, u, out);
}